// PlainVoxels_45457933861323
// MI455X (gfx1250) — compile-verified
//
#include <hip/hip_runtime.h>
#include <math.h>

// ---- problem constants (match reference) ----
#define GRID_R   256
#define CELLF    0.015f
#define ORIGINF  (-0.5f * GRID_R * CELLF)   // -1.92
#define SS       64
#define MIN_BETA 0.015f

#define RAYS_PER_BLOCK 16
#define THREADS (RAYS_PER_BLOCK * SS)       // 1024 = 32 waves (wave32)

typedef __attribute__((ext_vector_type(2))) float v2f;
typedef __attribute__((ext_vector_type(8))) float v8f;

__global__ __launch_bounds__(THREADS)
void plainvoxels_render_kernel(const float* __restrict__ grid,        // [256,256,256,5]
                               const float* __restrict__ beta,        // [1]
                               const float* __restrict__ rays_o,      // [N,3]
                               const float* __restrict__ rays_d,      // [N,3]
                               const float* __restrict__ rays_d_norm, // [N,1]
                               const float* __restrict__ nearp,       // [1]
                               const float* __restrict__ farp,        // [1]
                               float* __restrict__ out, int N)
{
    __shared__ float tau_lds[RAYS_PER_BLOCK][SS + 1]; // +1 pad: avoid 16-way bank conflict
    __shared__ float w_lds[RAYS_PER_BLOCK][SS + 1];
    __shared__ float acc_lds[RAYS_PER_BLOCK][8];      // r,g,b, w*t, nx,ny,nz, w

    const int tid  = threadIdx.x;
    const int lray = tid >> 6;        // local ray 0..15
    const int s    = tid & 63;        // sample 0..63
    const int ray  = blockIdx.x * RAYS_PER_BLOCK + lray;

    if (tid < RAYS_PER_BLOCK * 8) ((float*)acc_lds)[tid] = 0.0f;

    // ---------------- Phase 1: per-sample trilinear + gradient + tau ----------------
    const float nearv = nearp[0];
    const float farv  = farp[0];
    const float dt    = (farv - nearv) * (1.0f / (float)SS);
    const float t_mid = nearv + ((float)s + 0.5f) * dt;

    const float ox = rays_o[ray * 3 + 0], oy = rays_o[ray * 3 + 1], oz = rays_o[ray * 3 + 2];
    const float dxv = rays_d[ray * 3 + 0], dyv = rays_d[ray * 3 + 1], dzv = rays_d[ray * 3 + 2];

    const float px = ox + t_mid * dxv;
    const float py = oy + t_mid * dyv;
    const float pz = oz + t_mid * dzv;

    const float invc = 1.0f / CELLF;
    const float ux = (px - ORIGINF) * invc;
    const float uy = (py - ORIGINF) * invc;
    const float uz = (pz - ORIGINF) * invc;

    const bool mask = (ux >= 0.f) && (ux <= (float)(GRID_R - 1)) &&
                      (uy >= 0.f) && (uy <= (float)(GRID_R - 1)) &&
                      (uz >= 0.f) && (uz <= (float)(GRID_R - 1));

    int ix = (int)floorf(ux); ix = ix < 0 ? 0 : (ix > GRID_R - 2 ? GRID_R - 2 : ix);
    int iy = (int)floorf(uy); iy = iy < 0 ? 0 : (iy > GRID_R - 2 ? GRID_R - 2 : iy);
    int iz = (int)floorf(uz); iz = iz < 0 ? 0 : (iz > GRID_R - 2 ? GRID_R - 2 : iz);

    const float fx = ux - (float)ix;
    const float fy = uy - (float)iy;
    const float fz = uz - (float)iz;
    const float gxw = 1.f - fx, gyw = 1.f - fy, gzw = 1.f - fz;

    const size_t base = (((size_t)ix * GRID_R + (size_t)iy) * GRID_R + (size_t)iz) * 5;
    const size_t SX = (size_t)GRID_R * GRID_R * 5; // x stride
    const size_t SY = (size_t)GRID_R * 5;          // y stride

    float cf[8][4]; // corner cc = ddx*4 + ddy*2 + ddz, channels 0..3 (sdf, r, g, b)
#pragma unroll
    for (int cc = 0; cc < 8; ++cc) {
        const size_t off = base + (size_t)((cc >> 2) & 1) * SX
                                + (size_t)((cc >> 1) & 1) * SY
                                + (size_t)(cc & 1) * 5;
#pragma unroll
        for (int ch = 0; ch < 4; ++ch) cf[cc][ch] = grid[off + ch];
    }

    // trilinear weights
    const float w000 = gxw * gyw * gzw, w001 = gxw * gyw * fz;
    const float w010 = gxw * fy  * gzw, w011 = gxw * fy  * fz;
    const float w100 = fx  * gyw * gzw, w101 = fx  * gyw * fz;
    const float w110 = fx  * fy  * gzw, w111 = fx  * fy  * fz;

    float feat[4];
#pragma unroll
    for (int ch = 0; ch < 4; ++ch)
        feat[ch] = cf[0][ch]*w000 + cf[1][ch]*w001 + cf[2][ch]*w010 + cf[3][ch]*w011 +
                   cf[4][ch]*w100 + cf[5][ch]*w101 + cf[6][ch]*w110 + cf[7][ch]*w111;

    const float sdf = feat[0];
    const float c000 = cf[0][0], c001 = cf[1][0], c010 = cf[2][0], c011 = cf[3][0];
    const float c100 = cf[4][0], c101 = cf[5][0], c110 = cf[6][0], c111 = cf[7][0];

    // analytic gradient of trilinear interp (d/dx = (1/CELL) d/df)
    const float gx = (((c100 - c000) * gyw + (c110 - c010) * fy) * gzw +
                      ((c101 - c001) * gyw + (c111 - c011) * fy) * fz) * invc;
    const float gy = (((c010 - c000) * gxw + (c110 - c100) * fx) * gzw +
                      ((c011 - c001) * gxw + (c111 - c101) * fx) * fz) * invc;
    const float gz = (((c001 - c000) * gxw + (c101 - c100) * fx) * gyw +
                      ((c011 - c010) * gxw + (c111 - c110) * fx) * fy) * invc;

    // sdf_grads output: offset N*8, [N*S,3]
    {
        float* gout = out + (size_t)N * 8 + ((size_t)ray * SS + (size_t)s) * 3;
        gout[0] = gx; gout[1] = gy; gout[2] = gz;
    }

    const float gnorm = sqrtf(gx * gx + gy * gy + gz * gz);
    const float ginv  = 1.0f / fmaxf(gnorm, 1e-12f);
    const float nx = gx * ginv, ny = gy * ginv, nz = gz * ginv;

    // Laplace-CDF density
    const float beta_eff = MIN_BETA + fabsf(beta[0]);
    const float sgn = (sdf > 0.f) ? 1.f : ((sdf < 0.f) ? -1.f : 0.f);
    float sigma = (0.5f + 0.5f * sgn * expm1f(-fabsf(sdf) / beta_eff)) / beta_eff;
    sigma = mask ? sigma : 0.0f;
    const float tau = sigma * dt;

    tau_lds[lray][s] = tau;
    __syncthreads();

    // ---------------- Phase 2: WMMA prefix-sum over S, 4 waves in parallel ----------
    // CUM(16x64) = TAU(16x64) x U(64x64), U[k,n] = (k <= n), tiled as
    // V_WMMA_F32_16X16X4_F32 with triangular K range. Wave j (j=0..3) produces the
    // 16-sample output block j with 4*(j+1) chained WMMAs (max chain 16 vs 40 serial).
    // Each wave enters with all 32 lanes -> EXEC all ones, as WMMA requires.
    if (tid < 128) {
        const int j    = tid >> 5;   // wave id == output sample block
        const int lane = tid & 31;
        const int hf = lane >> 4;    // K/M half select per ISA lane layout
        const int nl = lane & 15;    // A: row M; B/C: column N
        v8f c = {};
        const int ncol = 16 * j + nl;            // global output column
        for (int kb = 0; kb <= 4 * j + 3; ++kb) {// K blocks of 4
            const int k0 = 4 * kb + 2 * hf;
            v2f a, b;
            a.x = tau_lds[nl][k0 + 0];           // A: M=nl, K=k0 / k0+1
            a.y = tau_lds[nl][k0 + 1];
            b.x = (k0 + 0) <= ncol ? 1.0f : 0.0f; // B: upper-triangular ones
            b.y = (k0 + 1) <= ncol ? 1.0f : 0.0f;
            c = __builtin_amdgcn_wmma_f32_16x16x4_f32(
                    /*neg_a=*/false, a, /*neg_b=*/false, b,
                    /*c_mod=*/(short)0, c, /*reuse_a=*/false, /*reuse_b=*/false);
        }
#pragma unroll
        for (int g = 0; g < 8; ++g) {            // C/D layout: row = g + 8*hf
            const int rr = g + 8 * hf;
            const int sc = 16 * j + nl;
            const float tv = tau_lds[rr][sc];
            const float T  = expf(tv - c[g]);    // exp(-(cumsum - tau))
            const float al = -expm1f(-tv);
            w_lds[rr][sc] = T * al;
        }
    }
    __syncthreads();

    // ---------------- Phase 3: per-ray weighted reductions ---------------------------
    const float wv = w_lds[lray][s];
    float vals[8] = { wv * feat[1], wv * feat[2], wv * feat[3], wv * t_mid,
                      wv * nx, wv * ny, wv * nz, wv };
#pragma unroll
    for (int ch = 0; ch < 8; ++ch) {
        float v = vals[ch];
#pragma unroll
        for (int o = 16; o > 0; o >>= 1) v += __shfl_xor(v, o, 32); // wave32: half a ray/wave
        if ((tid & 31) == 0) atomicAdd(&acc_lds[lray][ch], v);      // ds_add_f32
    }
    __syncthreads();

    if (tid < RAYS_PER_BLOCK) {
        const int rr = blockIdx.x * RAYS_PER_BLOCK + tid;
        const float inv_dn = 1.0f / rays_d_norm[rr];
        // rendered_rgb [N,3] @ 0
        out[(size_t)rr * 3 + 0] = acc_lds[tid][0];
        out[(size_t)rr * 3 + 1] = acc_lds[tid][1];
        out[(size_t)rr * 3 + 2] = acc_lds[tid][2];
        // rendered_depth [N,1] @ N*3
        out[(size_t)N * 3 + rr] = acc_lds[tid][3] * inv_dn;
        // rendered_normals [N,3] @ N*4
        out[(size_t)N * 4 + (size_t)rr * 3 + 0] = acc_lds[tid][4];
        out[(size_t)N * 4 + (size_t)rr * 3 + 1] = acc_lds[tid][5];
        out[(size_t)N * 4 + (size_t)rr * 3 + 2] = acc_lds[tid][6];
        // acc [N,1] @ N*7
        out[(size_t)N * 7 + rr] = acc_lds[tid][7];
        // near_out / far_out after sdf_grads (N*S*3) @ N*8
        const size_t tail = (size_t)N * 8 + (size_t)N * SS * 3;
        out[tail + rr]     = nearv * inv_dn;
        out[tail + N + rr] = farv  * inv_dn;
    }
}

extern "C" void kernel_launch(void* const* d_in, const int* in_sizes, int n_in,
                              void* d_out, int out_size, void* d_ws, size_t ws_size,
                              hipStream_t stream) {
    const float* grid        = (const float*)d_in[0];
    const float* beta        = (const float*)d_in[1];
    const float* rays_o      = (const float*)d_in[2];
    const float* rays_d      = (const float*)d_in[3];
    const float* rays_d_norm = (const float*)d_in[4];
    const float* nearp       = (const float*)d_in[5];
    const float* farp        = (const float*)d_in[6];
    const int N = in_sizes[2] / 3;                    // 4096
    const int blocks = N / RAYS_PER_BLOCK;            // 256
    plainvoxels_render_kernel<<<blocks, THREADS, 0, stream>>>(
        grid, beta, rays_o, rays_d, rays_d_norm, nearp, farp, (float*)d_out, N);
}